// TransposeTDLayer_70257075028073
// MI455X (gfx1250) — compile-verified
//
#include <hip/hip_runtime.h>
#include <hip/hip_bf16.h>

// Problem constants (from the reference)
#define B_     64
#define LIN_   512
#define CIN_   128
#define COUT_  128
#define K_     8
#define S_     2
#define OUT_   1030
// LDS row stride for staged x tile: CIN_ + 4 pad floats.
// 132 % 64(banks) == 4 -> 16 lanes at row-stride hit 16 distinct banks,
// and 132*4B = 528B keeps 16B alignment for b128 writes / 8B for b64 reads.
#define XS_STRIDE 132

typedef __attribute__((ext_vector_type(2))) float v2f;
typedef __attribute__((ext_vector_type(4))) float v4f;
typedef __attribute__((ext_vector_type(8))) float v8f;
typedef __attribute__((ext_vector_type(4))) int   v4i;

#if __has_builtin(__builtin_amdgcn_global_load_async_to_lds_b128) && \
    __has_builtin(__builtin_amdgcn_s_wait_asynccnt)
#define USE_ASYNC_LDS 1
#endif

__global__ __launch_bounds__(256) void ttd_wmma_f32_kernel(
    const float* __restrict__ x,     // (B, LIN, CIN)
    const float* __restrict__ w,     // (K, LIN, COUT, CIN)
    const float* __restrict__ bias,  // (OUT, COUT)
    float* __restrict__ y)           // (B, OUT, COUT)
{
    __shared__ float xs[B_ * XS_STRIDE];   // staged x[:, l, :], ~33 KB

    const int p     = blockIdx.x;      // output position, uniform per block
    const int tid   = threadIdx.x;
    const int lane  = tid & 31;
    const int wave  = tid >> 5;
    const int mrow  = lane & 15;       // A-matrix row within a 16-tile
    const int khalf = lane >> 4;       // 0/1: which K pair this lane holds
    const int kbase = khalf * 2;
    const int ncol  = wave * 16 + (lane & 15);  // output channel column

    // 4 M-tiles (batch rows 0..63) x this wave's 16 columns, f32 accum
    v8f acc[4] = {};

    // k must have the same parity as p: p = k + 2*l
    for (int k = (p & 1); k < K_; k += S_) {
        const int l = (p - k) >> 1;
        if (l < 0 || l >= LIN_) continue;       // uniform per block

        __syncthreads();                        // xs reuse protection

        // ---- Stage x[:, l, :] (64 x 128 f32) into padded LDS rows ----
#ifdef USE_ASYNC_LDS
        #pragma unroll
        for (int i = 0; i < 8; ++i) {
            const int idx = tid + (i << 8);     // 0..2047 b128 transfers
            const int row = idx >> 5;           // batch row 0..63
            const int seg = idx & 31;           // 16B segment within row
            const float* g = x + ((size_t)row * LIN_ + l) * CIN_ + seg * 4;
            float*       s = xs + row * XS_STRIDE + seg * 4;
            __builtin_amdgcn_global_load_async_to_lds_b128(
                (v4i*)g, (v4i*)s, 0, 0);
        }
        __builtin_amdgcn_s_wait_asynccnt(0);
#else
        #pragma unroll
        for (int i = 0; i < 8; ++i) {
            const int idx = tid + (i << 8);
            const int row = idx >> 5;
            const int seg = idx & 31;
            *(v4f*)(xs + row * XS_STRIDE + seg * 4) =
                *(const v4f*)(x + ((size_t)row * LIN_ + l) * CIN_ + seg * 4);
        }
#endif
        __syncthreads();

        // ---- GEMM-accumulate: acc += X_l (64x128) @ W[k,l]^T (128x128) ----
        // B-fragment source: W[k][l][ncol][c + kbase + {0,1}], streamed once.
        const float* wrow = w + (((size_t)k * LIN_ + l) * COUT_ + ncol) * CIN_ + kbase;
        const float* arow = xs + mrow * XS_STRIDE + kbase;
        __builtin_prefetch(wrow, 0, 0);         // global_prefetch_b8 for the row

        #pragma unroll 4
        for (int c = 0; c < CIN_; c += 4) {
            const v2f bf = *(const v2f*)(wrow + c);            // global b64
            #pragma unroll
            for (int mt = 0; mt < 4; ++mt) {
                const v2f af = *(const v2f*)(arow + mt * (16 * XS_STRIDE) + c); // ds b64
                acc[mt] = __builtin_amdgcn_wmma_f32_16x16x4_f32(
                    false, af, false, bf, (short)0, acc[mt], false, false);
            }
        }
    }

    // ---- Epilogue: bias + store. D layout: VGPR r -> M = khalf*8 + r ----
    const float bv = bias[(size_t)p * COUT_ + ncol];
    #pragma unroll
    for (int mt = 0; mt < 4; ++mt) {
        const int brow = mt * 16 + khalf * 8;
        #pragma unroll
        for (int r = 0; r < 8; ++r) {
            y[((size_t)(brow + r) * OUT_ + p) * COUT_ + ncol] = acc[mt][r] + bv;
        }
    }
}

extern "C" void kernel_launch(void* const* d_in, const int* in_sizes, int n_in,
                              void* d_out, int out_size, void* d_ws, size_t ws_size,
                              hipStream_t stream) {
    (void)in_sizes; (void)n_in; (void)out_size; (void)d_ws; (void)ws_size;
    const float* x    = (const float*)d_in[0];
    const float* w    = (const float*)d_in[1];
    const float* bias = (const float*)d_in[2];
    float*       y    = (float*)d_out;
    ttd_wmma_f32_kernel<<<dim3(OUT_), dim3(256), 0, stream>>>(x, w, bias, y);
}